// GnnMultipleInputFeatures_39908836114583
// MI455X (gfx1250) — compile-verified
//
#include <hip/hip_runtime.h>

typedef __attribute__((ext_vector_type(2))) float v2f;
typedef __attribute__((ext_vector_type(8))) float v8f;

// ---------------------------------------------------------------------------
// Workspace init: deg starts at 1.0 (self-loop contribution), accumulators 0.
// ---------------------------------------------------------------------------
__global__ void init_ws_kernel(float* __restrict__ deg,
                               float* __restrict__ agg1,
                               float* __restrict__ outacc, int N) {
    int i = blockIdx.x * blockDim.x + threadIdx.x;
    if (i < N * 16) agg1[i] = 0.0f;
    if (i < N) { deg[i] = 1.0f; outacc[i] = 0.0f; }
}

// deg[dst] += 1 per edge (self-loop already folded in via init=1.0)
__global__ void degree_kernel(const int* __restrict__ dst,
                              float* __restrict__ deg, int E) {
    int e = blockIdx.x * blockDim.x + threadIdx.x;
    if (e < E) atomicAdd(&deg[dst[e]], 1.0f);
}

// deg -> rsqrt(deg) in place (deg >= 1 always, keep guard for fidelity)
__global__ void dinv_kernel(float* __restrict__ deg, int N) {
    int i = blockIdx.x * blockDim.x + threadIdx.x;
    if (i < N) {
        float d = deg[i];
        deg[i] = (d > 0.0f) ? rsqrtf(d) : 0.0f;
    }
}

// ---------------------------------------------------------------------------
// Layer-1 linear via V_WMMA_F32_16X16X4_F32.
// One wave32 computes a 16-node x 16-feature tile:
//   A[16x4]: A[m][0] = 2*x[m,0], A[m][1] = x[m,2], K=2..3 zero-padded.
//     lanes 0-15 hold (K0,K1) for M=lane; lanes 16-31 hold (K2,K3)=0.
//   B[4x16]: B[0][n]=W1[0][n], B[1][n]=W1[1][n]; rows 2,3 zero.
//   D[16x16]: VGPR v -> M=v (lanes 0-15) / M=v+8 (lanes 16-31), N=lane%16.
// EXEC kept all-ones across the WMMA (wave-uniform exit; lane masking via
// clamped indices + selects, no divergent branches before the WMMA).
// Store path: tile-fullness is wave-uniform -> fast path of 8 unconditional
// clause-able stores; masked tail only for a partial final tile.
// ---------------------------------------------------------------------------
__global__ void linear1_wmma_kernel(const float* __restrict__ x,   // [N,3]
                                    const float* __restrict__ W1,  // [2,16]
                                    float* __restrict__ h1,        // [N,16]
                                    int N) {
    int lane = threadIdx.x & 31;
    int wave = (blockIdx.x * blockDim.x + threadIdx.x) >> 5;
    int base = wave * 16;
    if (base >= N) return;                 // wave-uniform: whole wave exits

    int half = lane >> 4;                  // 0: K0/K1 lanes, 1: K2/K3 (pad)
    int idx  = lane & 15;                  // M for A-lanes, N for B/D
    int node  = base + idx;
    int nodeC = node < N ? node : (N - 1); // clamped, non-divergent load
    float aval = (half == 0 && node < N) ? 1.0f : 0.0f;

    v2f a;
    a.x = aval * 2.0f * x[nodeC * 3 + 0];  // feature 0, scaled by 2
    a.y = aval *        x[nodeC * 3 + 2];  // feature 2

    float bval = (half == 0) ? 1.0f : 0.0f;
    v2f b;
    b.x = bval * W1[idx];                  // W1[0][idx]
    b.y = bval * W1[16 + idx];             // W1[1][idx]

    v8f c = {};
    c = __builtin_amdgcn_wmma_f32_16x16x4_f32(
            /*neg_a=*/false, a, /*neg_b=*/false, b,
            /*c_mod=*/(short)0, c, /*reuse_a=*/false, /*reuse_b=*/false);

    // This lane owns column `idx` of rows [base+half*8, base+half*8+8).
    float* p = h1 + ((size_t)(base + half * 8) * 16 + idx);
    if (base + 16 <= N) {                  // wave-uniform fast path
#pragma unroll
        for (int v = 0; v < 8; ++v) p[v * 16] = c[v];
    } else {                               // partial final tile (rare)
#pragma unroll
        for (int v = 0; v < 8; ++v) {
            if (base + half * 8 + v < N) p[v * 16] = c[v];
        }
    }
}

// ---------------------------------------------------------------------------
// Layer-1 aggregation: 4 threads per edge, each handles a float4 feature
// chunk.  agg1[dst] += h1[src] * dinv[src]*dinv[dst]  (16-wide atomics).
// All node arrays (h1/agg1/dinv ~13 MB) are L2-resident on MI455X (192 MB L2)
// so the gathers and atomics resolve in L2; the sequential edge-list read is
// prefetched ahead.
// ---------------------------------------------------------------------------
__global__ void agg1_kernel(const int* __restrict__ src,
                            const int* __restrict__ dst,
                            const float* __restrict__ dinv,
                            const float* __restrict__ h1,
                            float* __restrict__ agg1, int E) {
    int t = blockIdx.x * blockDim.x + threadIdx.x;
    int e = t >> 2;
    int q = t & 3;
    if (e >= E) return;
    __builtin_prefetch(&src[e + 2048], 0, 0);   // global_prefetch_b8 (stream)
    __builtin_prefetch(&dst[e + 2048], 0, 0);
    int s = src[e];
    int d = dst[e];
    float norm = dinv[s] * dinv[d];
    float4 hv = ((const float4*)(h1 + (size_t)s * 16))[q];  // 64B-aligned rows
    float* ap = agg1 + (size_t)d * 16 + q * 4;
    atomicAdd(ap + 0, hv.x * norm);
    atomicAdd(ap + 1, hv.y * norm);
    atomicAdd(ap + 2, hv.z * norm);
    atomicAdd(ap + 3, hv.w * norm);
}

// ---------------------------------------------------------------------------
// Fused: add self-loop term + bias, ReLU, then layer-2 linear (16 -> 1).
//   z[n] = sum_j relu(agg1[n,j] + h1[n,j]*dinv[n]^2 + b1[j]) * W3[j]
// ---------------------------------------------------------------------------
__global__ void fuse_relu_linear2_kernel(const float* __restrict__ agg1,
                                         const float* __restrict__ h1,
                                         const float* __restrict__ dinv,
                                         const float* __restrict__ b1,
                                         const float* __restrict__ W3,
                                         float* __restrict__ z, int N) {
    int n = blockIdx.x * blockDim.x + threadIdx.x;
    if (n >= N) return;
    float di = dinv[n];
    float selfn = di * di;
    float acc = 0.0f;
#pragma unroll
    for (int j = 0; j < 16; ++j) {
        float v = agg1[(size_t)n * 16 + j] + h1[(size_t)n * 16 + j] * selfn + b1[j];
        v = fmaxf(v, 0.0f);
        acc = fmaf(v, W3[j], acc);
    }
    z[n] = acc;
}

// Layer-2 aggregation: scalar feature per node.
__global__ void agg2_kernel(const int* __restrict__ src,
                            const int* __restrict__ dst,
                            const float* __restrict__ dinv,
                            const float* __restrict__ z,
                            float* __restrict__ outacc, int E) {
    int e = blockIdx.x * blockDim.x + threadIdx.x;
    if (e >= E) return;
    __builtin_prefetch(&src[e + 8192], 0, 0);
    __builtin_prefetch(&dst[e + 8192], 0, 0);
    int s = src[e];
    int d = dst[e];
    atomicAdd(&outacc[d], z[s] * dinv[s] * dinv[d]);
}

// out[n] = outacc[n] + z[n]*dinv[n]^2 + b3
__global__ void finalize_kernel(const float* __restrict__ outacc,
                                const float* __restrict__ z,
                                const float* __restrict__ dinv,
                                const float* __restrict__ b3,
                                float* __restrict__ out, int N) {
    int n = blockIdx.x * blockDim.x + threadIdx.x;
    if (n >= N) return;
    float di = dinv[n];
    out[n] = outacc[n] + z[n] * di * di + b3[0];
}

extern "C" void kernel_launch(void* const* d_in, const int* in_sizes, int n_in,
                              void* d_out, int out_size, void* d_ws, size_t ws_size,
                              hipStream_t stream) {
    const float* x   = (const float*)d_in[0];  // [N,3]
    const float* W1  = (const float*)d_in[1];  // [2,16]
    const float* b1  = (const float*)d_in[2];  // [16]
    const float* W3  = (const float*)d_in[3];  // [16,1]
    const float* b3  = (const float*)d_in[4];  // [1]
    const int* edges = (const int*)d_in[5];    // [2,E] (harness: integer -> int32)

    const int N = in_sizes[0] / 3;
    const int E = in_sizes[5] / 2;
    const int* src = edges;       // edge_index[0]
    const int* dst = edges + E;   // edge_index[1]

    // Workspace layout (floats): dinv[N] | z[N] | outacc[N] | h1[16N] | agg1[16N]
    float* ws     = (float*)d_ws;
    float* dinv   = ws;
    float* z      = ws + (size_t)N;
    float* outacc = ws + (size_t)2 * N;
    float* h1     = ws + (size_t)3 * N;
    float* agg1   = ws + (size_t)19 * N;

    const int T = 256;
    int gInit = (N * 16 + T - 1) / T;
    init_ws_kernel<<<gInit, T, 0, stream>>>(dinv, agg1, outacc, N);

    int gE = (E + T - 1) / T;
    degree_kernel<<<gE, T, 0, stream>>>(dst, dinv, E);

    int gN = (N + T - 1) / T;
    dinv_kernel<<<gN, T, 0, stream>>>(dinv, N);

    int waves = (N + 15) / 16;                    // one wave32 per 16-node tile
    int gW = (waves * 32 + T - 1) / T;
    linear1_wmma_kernel<<<gW, T, 0, stream>>>(x, W1, h1, N);

    int gA1 = (E * 4 + T - 1) / T;                // 4 threads per edge
    agg1_kernel<<<gA1, T, 0, stream>>>(src, dst, dinv, h1, agg1, E);

    fuse_relu_linear2_kernel<<<gN, T, 0, stream>>>(agg1, h1, dinv, b1, W3, z, N);

    agg2_kernel<<<gE, T, 0, stream>>>(src, dst, dinv, z, outacc, E);

    finalize_kernel<<<gN, T, 0, stream>>>(outacc, z, dinv, b3, (float*)d_out, N);
}